// ROIPooling_37615323578920
// MI455X (gfx1250) — compile-verified
//
#include <hip/hip_runtime.h>
#include <stdint.h>

// ---------------------------------------------------------------------------
// ROI-Align (crop_and_resize bilinear, 14x14 samples) + 2x2 per-bin max pool.
//   x:    [8, 64, 64, 256]  f32 (NHWC, C innermost)
//   rois: [8, 300, 4]       f32 (y1,x1,y2,x2 normalized)
//   out:  [8, 300, 7, 7, 256] f32
//
// Roofline: ~1 GFLOP of bilinear blends vs 33.5 MB input (L2-resident) +
// 120 MB write-once output -> HBM floor ~6.6us @ 23.3 TB/s. Pure gather /
// data-movement problem; no dense-matmul structure -> WMMA inapplicable.
// CDNA5 path: TDM gather-mode tensor_load_to_lds (16x 16-bit row indices =
// the 16 bilinear tap pixels of one bin, rows of a [H*W, C] tensor),
// double-buffered across the 7 bins of a row, TENSORcnt waits, NT stores.
// One wave32 block per (b, roi, ky); 32 KB LDS -> 10 blocks/WGP.
// ---------------------------------------------------------------------------

typedef unsigned int u32;
typedef u32   v4u __attribute__((ext_vector_type(4)));
typedef int   v8i __attribute__((ext_vector_type(8)));
typedef int   v4i __attribute__((ext_vector_type(4)));
typedef float f4  __attribute__((ext_vector_type(4)));

#define KK   7
#define SUBS 2
#define BB   8
#define HH   64
#define WW   64
#define CC   256
#define NN   300

__device__ __forceinline__ int rfl(int v) { return __builtin_amdgcn_readfirstlane(v); }

__device__ __forceinline__ f4 f4max(f4 a, f4 b) {
  f4 r;
  r.x = fmaxf(a.x, b.x); r.y = fmaxf(a.y, b.y);
  r.z = fmaxf(a.z, b.z); r.w = fmaxf(a.w, b.w);
  return r;
}

// Issue one TDM gather: 16 rows (16-bit indices into a [H*W, 256] f32 tensor)
// -> 16 KB contiguous in LDS at lds_off. Descriptor per CDNA5 ISA.
__device__ __forceinline__ void tdm_issue(uint64_t gaddr, u32 lds_off, const u32* p) {
  // Group 0: count=1, gather_mode=1 (bit31), 16-bit indices (bit30=0),
  //          lds_addr[63:32], global_addr[120:64], type=2 (bits 127:126).
  v4u g0 = { (u32)rfl((int)(1u | (1u << 31))),
             (u32)rfl((int)lds_off),
             (u32)rfl((int)(u32)gaddr),
             (u32)rfl((int)((u32)(gaddr >> 32) | (2u << 30))) };
  // Group 1: wg_mask=0, data_size=2 (4B), tensor_dim0=256, tensor_dim1=4096,
  //          tile_dim0=256, tile_dim1=16 (#gather indices), dim0_stride=256.
  v8i g1 = { rfl(2 << 16),        // [17:16] data_size = 4 bytes
             rfl(256 << 16),      // tensor_dim0[15:0] at bits 63:48
             rfl(4096 << 16),     // tensor_dim1[15:0] at bits 95:80
             rfl(256 << 16),      // tile_dim0 at bits 127:112
             rfl(16),             // tile_dim1 = 16 valid gather indices
             rfl(256),            // tensor_dim0_stride[31:0]
             0, 0 };
  // Groups 2/3: 16x 16-bit row indices, 2 per dword.
  v4i g2 = { rfl((int)p[0]), rfl((int)p[1]), rfl((int)p[2]), rfl((int)p[3]) };
  v4i g3 = { rfl((int)p[4]), rfl((int)p[5]), rfl((int)p[6]), rfl((int)p[7]) };
#if defined(__clang_major__) && (__clang_major__ >= 23)
  v8i gz = {0, 0, 0, 0, 0, 0, 0, 0};
  __builtin_amdgcn_tensor_load_to_lds(g0, g1, g2, g3, gz, 0);
#else
  __builtin_amdgcn_tensor_load_to_lds(g0, g1, g2, g3, 0);
#endif
}

__global__ __launch_bounds__(32) void roialign_maxpool_kernel(
    const float* __restrict__ x, const float* __restrict__ rois,
    float* __restrict__ out) {
  __shared__ float tile[2 * 16 * CC];   // double buffer, 2 x 16 KB

  const int ky  = blockIdx.x;           // 0..6
  const int n   = blockIdx.y;           // 0..299
  const int b   = blockIdx.z;           // 0..7
  const int bn  = b * NN + n;
  const int tid = threadIdx.x;          // 0..31, one wave32

  const float ry1 = rois[bn * 4 + 0];
  const float rx1 = rois[bn * 4 + 1];
  const float ry2 = rois[bn * 4 + 2];
  const float rx2 = rois[bn * 4 + 3];

  // Vertical sample positions for this bin row (shared by all 7 bins).
  int   y0v[SUBS];
  float wyv[SUBS];
#pragma unroll
  for (int sy = 0; sy < SUBS; ++sy) {
    float t  = (float)(ky * SUBS + sy) * (1.0f / (KK * SUBS - 1));
    float ys = fminf(fmaxf((ry1 + t * (ry2 - ry1)) * (HH - 1), 0.0f), (float)(HH - 1));
    int   y0 = (int)floorf(ys);
    y0 = y0 > HH - 2 ? HH - 2 : y0;
    y0v[sy] = y0;
    wyv[sy] = ys - (float)y0;
  }

  const uint64_t gaddr    = (uint64_t)(uintptr_t)(x + (size_t)b * HH * WW * CC);
  const u32      lds_base = (u32)(size_t)(&tile[0]);

  auto xpos = [&](int kx, int* x0v, float* wxv) {
#pragma unroll
    for (int sx = 0; sx < SUBS; ++sx) {
      float t  = (float)(kx * SUBS + sx) * (1.0f / (KK * SUBS - 1));
      float xs = fminf(fmaxf((rx1 + t * (rx2 - rx1)) * (WW - 1), 0.0f), (float)(WW - 1));
      int   x0 = (int)floorf(xs);
      x0 = x0 > WW - 2 ? WW - 2 : x0;
      x0v[sx] = x0;
      wxv[sx] = xs - (float)x0;
    }
  };

  auto issue = [&](int kx, int buf) {
    int x0v[SUBS]; float wxv[SUBS];
    xpos(kx, x0v, wxv);
    u32 p[8];   // 16 row indices, slot q = (sy*2+sx)*4 + tap
#pragma unroll
    for (int s = 0; s < 4; ++s) {
      int sy = s >> 1, sx = s & 1;
      u32 r00 = (u32)((y0v[sy]    ) * WW + x0v[sx]    );
      u32 r01 = (u32)((y0v[sy]    ) * WW + x0v[sx] + 1);
      u32 r10 = (u32)((y0v[sy] + 1) * WW + x0v[sx]    );
      u32 r11 = (u32)((y0v[sy] + 1) * WW + x0v[sx] + 1);
      p[s * 2 + 0] = r00 | (r01 << 16);
      p[s * 2 + 1] = r10 | (r11 << 16);
    }
    tdm_issue(gaddr, lds_base + (u32)buf * 16u * CC * 4u, p);
  };

  auto compute = [&](int kx, int buf) {
    int x0v[SUBS]; float wxv[SUBS];
    xpos(kx, x0v, wxv);
    const f4* tp = (const f4*)(&tile[buf * 16 * CC]);
    f4 acc0, acc1;
#pragma unroll
    for (int s = 0; s < 4; ++s) {
      int   sy = s >> 1, sx = s & 1;
      float wy = wyv[sy], wx = wxv[sx];
      float w00 = (1.0f - wy) * (1.0f - wx);
      float w01 = (1.0f - wy) * wx;
      float w10 = wy * (1.0f - wx);
      float w11 = wy * wx;
      int base = (s * 4) * (CC / 4) + tid * 2;   // row q = s*4+tap, 64 f4 per row
      f4 a0 = tp[base + 0 * 64 + 0], a1 = tp[base + 0 * 64 + 1];
      f4 b0 = tp[base + 1 * 64 + 0], b1 = tp[base + 1 * 64 + 1];
      f4 c0 = tp[base + 2 * 64 + 0], c1 = tp[base + 2 * 64 + 1];
      f4 d0 = tp[base + 3 * 64 + 0], d1 = tp[base + 3 * 64 + 1];
      f4 v0 = a0 * w00 + b0 * w01 + c0 * w10 + d0 * w11;
      f4 v1 = a1 * w00 + b1 * w01 + c1 * w10 + d1 * w11;
      if (s == 0) { acc0 = v0; acc1 = v1; }
      else        { acc0 = f4max(acc0, v0); acc1 = f4max(acc1, v1); }
    }
    // out[b,n,ky,kx,c], c = tid*8 .. tid*8+7; write-once -> non-temporal.
    float* op = out + (((size_t)bn * (KK * KK)) + (size_t)(ky * KK + kx)) * CC + tid * 8;
    __builtin_nontemporal_store(acc0, (f4*)(op));
    __builtin_nontemporal_store(acc1, (f4*)(op + 4));
  };

  issue(0, 0);
#pragma unroll
  for (int kx = 0; kx < KK; ++kx) {
    if (kx + 1 < KK) {
      issue(kx + 1, (kx + 1) & 1);
      __builtin_amdgcn_s_wait_tensorcnt(1);   // current buffer done (in-order TDM)
    } else {
      __builtin_amdgcn_s_wait_tensorcnt(0);
    }
    asm volatile("" ::: "memory");            // LDS was written by TDM, not VALU
    compute(kx, kx & 1);
  }
}

extern "C" void kernel_launch(void* const* d_in, const int* in_sizes, int n_in,
                              void* d_out, int out_size, void* d_ws, size_t ws_size,
                              hipStream_t stream) {
  (void)in_sizes; (void)n_in; (void)out_size; (void)d_ws; (void)ws_size;
  const float* x    = (const float*)d_in[0];   // [8,64,64,256] f32
  const float* rois = (const float*)d_in[1];   // [8,300,4]     f32
  float* out = (float*)d_out;                  // [8,300,7,7,256] f32

  dim3 grid(KK, NN, BB);   // ky, n, b -> no integer division in the kernel
  roialign_maxpool_kernel<<<grid, dim3(32, 1, 1), 0, stream>>>(x, rois, out);
}